// MoeLayer_56727928045674
// MI455X (gfx1250) — compile-verified
//
#include <hip/hip_runtime.h>

#define N_    64
#define HW_   4096
#define C_    128
#define E_    8

typedef __attribute__((ext_vector_type(2))) float v2f;
typedef __attribute__((ext_vector_type(4))) float v4f;
typedef __attribute__((ext_vector_type(8))) float v8f;

// ---------------------------------------------------------------- phase 0: zero pooled
__global__ void zero_pooled_kernel(float* __restrict__ pooled) {
    int i = blockIdx.x * blockDim.x + threadIdx.x;
    if (i < N_ * C_) pooled[i] = 0.0f;
}

// ---------------------------------------------------------------- phase 1: global avg pool (raw sums)
// grid = 64 samples * 64 chunks, block = 256. Chunk = 64 pixels.
__global__ __launch_bounds__(256) void pool_kernel(const float* __restrict__ x,
                                                   float* __restrict__ pooled) {
    int bx    = blockIdx.x;
    int n     = bx >> 6;
    int chunk = bx & 63;
    int t     = threadIdx.x;
    int c     = t & 127;
    int pr    = t >> 7;                       // 0..1
    const float* xp = x + ((size_t)n * HW_ + (size_t)chunk * 64) * C_;
    float s = 0.0f;
#pragma unroll 8
    for (int i = pr; i < 64; i += 2) s += xp[i * C_ + c];
    __shared__ float red[256];
    red[t] = s;
    __syncthreads();
    if (t < 128) atomicAdd(&pooled[n * C_ + c], red[t] + red[t + 128]);
}

// ---------------------------------------------------------------- phase 2: gating (1 block)
__global__ __launch_bounds__(256) void gate_kernel(const float* __restrict__ pooled,
                                                   const float* __restrict__ gate_w,
                                                   const float* __restrict__ gate_b,
                                                   const float* __restrict__ expert_b,
                                                   float* __restrict__ wsel,
                                                   int*   __restrict__ esel,
                                                   float* __restrict__ bcomb,
                                                   float* __restrict__ laux_out) {
    __shared__ float gbuf[N_][E_];
    __shared__ float wbuf[N_][2];
    __shared__ int   sbuf[N_][2];
    __shared__ int   amax[N_];
    __shared__ float part[E_];
    int t = threadIdx.x;

    // logits (pooled holds raw sums; fold in 1/4096)
    for (int idx = t; idx < N_ * E_; idx += 256) {
        int n = idx >> 3, e = idx & 7;
        float acc = 0.0f;
        for (int c = 0; c < C_; ++c) acc += pooled[n * C_ + c] * gate_w[c * E_ + e];
        gbuf[n][e] = acc * (1.0f / 4096.0f) + gate_b[e];
    }
    __syncthreads();

    if (t < N_) {
        float v[E_];
        float mx = -1e30f;
        for (int e = 0; e < E_; ++e) { v[e] = gbuf[t][e]; mx = fmaxf(mx, v[e]); }
        float s = 0.0f;
        for (int e = 0; e < E_; ++e) { v[e] = __expf(v[e] - mx); s += v[e]; }
        float inv = 1.0f / s;
        int i0 = 0, i1 = 0; float v0 = -1e30f, v1 = -1e30f;
        for (int e = 0; e < E_; ++e) {
            float g = v[e] * inv;
            gbuf[t][e] = g;
            if (g > v0)      { v1 = v0; i1 = i0; v0 = g; i0 = e; }
            else if (g > v1) { v1 = g;  i1 = e; }
        }
        amax[t] = i0;
        // renormalize top-2 via softmax (max-subtracted, exact)
        float e1 = __expf(v1 - v0);
        float w0 = 1.0f / (1.0f + e1);
        float w1 = e1 * w0;
        wbuf[t][0] = w0; wbuf[t][1] = w1;
        sbuf[t][0] = i0; sbuf[t][1] = i1;
        wsel[t * 2 + 0] = w0; wsel[t * 2 + 1] = w1;
        esel[t * 2 + 0] = i0; esel[t * 2 + 1] = i1;
    }
    __syncthreads();

    if (t < E_) {
        float me = 0.0f, ce = 0.0f;
        for (int n = 0; n < N_; ++n) {
            me += gbuf[n][t];
            ce += (amax[n] == t) ? 1.0f : 0.0f;
        }
        part[t] = (me * (1.0f / 64.0f)) * (ce * (1.0f / 64.0f));
    }
    __syncthreads();
    if (t == 0) {
        float s = 0.0f;
        for (int e = 0; e < E_; ++e) s += part[e];
        *laux_out = s * 8.0f;   // mean over E (1/8) * E*E (64)
    }

    // combined bias
    for (int idx = t; idx < N_ * C_; idx += 256) {
        int n = idx >> 7, o = idx & 127;
        bcomb[idx] = wbuf[n][0] * expert_b[sbuf[n][0] * C_ + o]
                   + wbuf[n][1] * expert_b[sbuf[n][1] * C_ + o];
    }
}

// ---------------------------------------------------------------- phase 3: combined expert weights
// grid = 64 samples * 16 chunks, block = 256, float4 per thread (1024 floats / chunk)
__global__ __launch_bounds__(256) void combine_kernel(const float* __restrict__ expert_w,
                                                      const float* __restrict__ wsel,
                                                      const int*   __restrict__ esel,
                                                      float* __restrict__ wcomb) {
    int bx = blockIdx.x;
    int n = bx >> 4, chunk = bx & 15;
    float w0 = wsel[n * 2 + 0], w1 = wsel[n * 2 + 1];
    int   e0 = esel[n * 2 + 0], e1 = esel[n * 2 + 1];
    int t = threadIdx.x;
    const v4f* A = (const v4f*)(expert_w + (size_t)e0 * 16384 + chunk * 1024);
    const v4f* B = (const v4f*)(expert_w + (size_t)e1 * 16384 + chunk * 1024);
    v4f*       D = (v4f*)(wcomb + (size_t)n * 16384 + chunk * 1024);
    D[t] = w0 * A[t] + w1 * B[t];
}

// ---------------------------------------------------------------- phase 4: main per-sample GEMM (WMMA f32)
// grid = 64 samples * 256 M-tiles, block = 256 (8 wave32s).
// Block: stages 16 px x 128 ch of X into LDS; wave w computes the 16x16 D tile
// for output channels [16w, 16w+16) with 32 x v_wmma_f32_16x16x4_f32 over K=128.
__global__ __launch_bounds__(256) void moe_gemm_kernel(const float* __restrict__ x,
                                                       const float* __restrict__ wcomb,
                                                       const float* __restrict__ bcomb,
                                                       float* __restrict__ out) {
    int bx    = blockIdx.x;
    int n     = bx >> 8;
    int mtile = bx & 255;
    int p0    = mtile * 16;
    int t     = threadIdx.x;

    // row stride 132 floats -> bank-conflict-free b64 A-fragment reads (132 % 64 == 4)
    __shared__ float xs[16 * 132];
    {
        int r  = t >> 4;
        int c8 = (t & 15) * 8;
        const float* src = x + ((size_t)(n * HW_ + p0 + r)) * C_ + c8;
        v4f a0 = *(const v4f*)(src);
        v4f a1 = *(const v4f*)(src + 4);
        *(v4f*)&xs[r * 132 + c8]     = a0;
        *(v4f*)&xs[r * 132 + c8 + 4] = a1;
    }
    __syncthreads();

    int w    = t >> 5;
    int lane = t & 31;
    int m    = lane & 15;            // A row (pixel) / B row (out-ch) / D col index
    int kb   = (lane >> 4) * 2;      // K pair selected by lane half
    int o0   = w * 16;
    int ocol = o0 + m;

    float bias = bcomb[n * C_ + ocol];
    v8f c;
#pragma unroll
    for (int j = 0; j < 8; ++j) c[j] = bias;

    const float* wrow = wcomb + ((size_t)n * C_ + ocol) * C_ + kb;   // W_comb[n][ocol][kb + k0]
    const float* xrow = &xs[m * 132 + kb];

#pragma unroll
    for (int k0 = 0; k0 < C_; k0 += 4) {
        v2f a = *(const v2f*)(xrow + k0);   // A: X[p0+m][k0+kb .. +1]
        v2f b = *(const v2f*)(wrow + k0);   // B = W^T: W[ocol][k0+kb .. +1]
        c = __builtin_amdgcn_wmma_f32_16x16x4_f32(false, a, false, b, (short)0, c,
                                                  false, false);
    }

    // D layout: VGPR j, lanes 0-15: M=j, N=lane; lanes 16-31: M=j+8, N=lane-16
    float* orow = out + ((size_t)(n * HW_ + p0)) * C_ + o0 + m;
    int mbase = (lane >> 4) * 8;
#pragma unroll
    for (int j = 0; j < 8; ++j) orow[(size_t)(mbase + j) * C_] = c[j];
}

// ---------------------------------------------------------------- launch
extern "C" void kernel_launch(void* const* d_in, const int* in_sizes, int n_in,
                              void* d_out, int out_size, void* d_ws, size_t ws_size,
                              hipStream_t stream) {
    const float* x        = (const float*)d_in[0];
    const float* gate_w   = (const float*)d_in[1];
    const float* gate_b   = (const float*)d_in[2];
    const float* expert_w = (const float*)d_in[3];
    const float* expert_b = (const float*)d_in[4];
    float* out = (float*)d_out;

    char*  ws     = (char*)d_ws;
    float* pooled = (float*)(ws);            // 8192 f  (raw sums)
    float* wsel   = (float*)(ws + 32768);    // 128 f
    int*   esel   = (int*)  (ws + 33280);    // 128 i
    float* bcomb  = (float*)(ws + 33792);    // 8192 f
    float* wcomb  = (float*)(ws + 66560);    // 1,048,576 f (16B aligned)

    float* laux = out + (size_t)N_ * HW_ * C_;   // tuple output: results flat, then l_aux

    zero_pooled_kernel<<<(N_ * C_ + 255) / 256, 256, 0, stream>>>(pooled);
    pool_kernel<<<N_ * 64, 256, 0, stream>>>(x, pooled);
    gate_kernel<<<1, 256, 0, stream>>>(pooled, gate_w, gate_b, expert_b,
                                       wsel, esel, bcomb, laux);
    combine_kernel<<<N_ * 16, 256, 0, stream>>>(expert_w, wsel, esel, wcomb);
    moe_gemm_kernel<<<N_ * 256, 256, 0, stream>>>(x, wcomb, bcomb, out);
}